// GAT_30039001268561
// MI455X (gfx1250) — compile-verified
//
#include <hip/hip_runtime.h>
#include <math.h>

typedef float v2f __attribute__((ext_vector_type(2)));
typedef float v8f __attribute__((ext_vector_type(8)));

#define NEG_ATT 0.2f    // GATConv internal leaky slope
#define NEG_OUT 0.01f   // jax.nn.leaky_relu default slope
#define NGRAPH 256

// ---------------- utility ----------------
__global__ void __launch_bounds__(256) k_fill(float* p, int n, float v) {
  int i = blockIdx.x * 256 + threadIdx.x;
  if (i < n) p[i] = v;
}

__device__ __forceinline__ void atomicMaxF(float* addr, float v) {
  // sign-split trick: works for mixed-sign IEEE floats with -inf init
  if (__float_as_int(v) >= 0) atomicMax((int*)addr, __float_as_int(v));
  else                        atomicMin((unsigned int*)addr, __float_as_uint(v));
}

// ---------------- WMMA GEMM: Hout[N,64] = X[N,K] @ W[K,64] ----------------
// block = 256 threads = 8 waves; block tile = 32 rows x 64 cols; wave tile 16x16.
__global__ void __launch_bounds__(256) k_gemm_wmma(const float* __restrict__ X,
                                                   const float* __restrict__ W,
                                                   float* __restrict__ Hout,
                                                   int N, int K) {
  __shared__ float Wl[128 * 64];               // up to 32 KB (K<=128)
  const int tid = threadIdx.x;
  for (int i = tid; i < K * 64; i += 256) Wl[i] = W[i];
  __syncthreads();

  const int wave = tid >> 5;
  const int lane = tid & 31;
  const int g  = lane >> 4;                    // half-wave
  const int li = lane & 15;
  const int row0 = blockIdx.x * 32 + (wave >> 2) * 16;
  const int col0 = (wave & 3) * 16;

  const int ra = min(row0 + li, N - 1);        // clamp A-row loads (EXEC stays all-1)
  const float* __restrict__ xrow = X + (size_t)ra * K;

  v8f acc = {};
  for (int kk = 0; kk < K; kk += 4) {
    const int k0 = kk + 2 * g;                 // A: lane holds (M=li, K=k0..k0+1)
    v2f a, b;
    a.x = xrow[k0];
    a.y = xrow[k0 + 1];
    b.x = Wl[k0 * 64 + col0 + li];             // B: lane holds (K=k0..k0+1, N=li)
    b.y = Wl[(k0 + 1) * 64 + col0 + li];
    acc = __builtin_amdgcn_wmma_f32_16x16x4_f32(false, a, false, b, (short)0, acc,
                                                false, false);
  }
#pragma unroll
  for (int r = 0; r < 8; ++r) {                // D: vgpr r -> (M=r+8g, N=li)
    int row = row0 + r + 8 * g;
    if (row < N) Hout[(size_t)row * 64 + col0 + li] = acc[r];
  }
}

// ---------------- per-(node,head) attention scores ----------------
__global__ void __launch_bounds__(256) k_attn_scores(const float* __restrict__ Hn,
                                                     const float* __restrict__ as,
                                                     const float* __restrict__ ad,
                                                     float* __restrict__ ssrc,
                                                     float* __restrict__ sdst,
                                                     int N, int heads, int C) {
  int idx = blockIdx.x * 256 + threadIdx.x;
  if (idx >= N * heads) return;
  int n = idx / heads, hh = idx - n * heads;
  const float* hp = Hn + (size_t)n * 64 + hh * C;
  float s1 = 0.f, s2 = 0.f;
  for (int c = 0; c < C; ++c) {
    float v = hp[c];
    s1 += v * as[hh * C + c];
    s2 += v * ad[hh * C + c];
  }
  ssrc[idx] = s1;
  sdst[idx] = s2;
}

// ---------------- edge kernels (self-loops appended: e>=E -> src=dst=e-E) ----
__device__ __forceinline__ void edge_ends(const int* ei, int E, int e, int& s, int& d) {
  if (e < E) { s = ei[e]; d = ei[E + e]; } else { s = d = e - E; }
}

__global__ void __launch_bounds__(256) k_edge_max(const int* __restrict__ ei, int E, int Etot,
                                                  const float* __restrict__ ssrc,
                                                  const float* __restrict__ sdst,
                                                  float* __restrict__ mx, int heads) {
  int idx = blockIdx.x * 256 + threadIdx.x;
  if (idx >= Etot * heads) return;
  int e = idx / heads, hh = idx - e * heads;
  int s, d; edge_ends(ei, E, e, s, d);
  float v = ssrc[s * heads + hh] + sdst[d * heads + hh];
  v = v > 0.f ? v : NEG_ATT * v;
  atomicMaxF(&mx[d * heads + hh], v);
}

__global__ void __launch_bounds__(256) k_edge_sum(const int* __restrict__ ei, int E, int Etot,
                                                  const float* __restrict__ ssrc,
                                                  const float* __restrict__ sdst,
                                                  const float* __restrict__ mx,
                                                  float* __restrict__ den, int heads) {
  int idx = blockIdx.x * 256 + threadIdx.x;
  if (idx >= Etot * heads) return;
  int e = idx / heads, hh = idx - e * heads;
  int s, d; edge_ends(ei, E, e, s, d);
  float v = ssrc[s * heads + hh] + sdst[d * heads + hh];
  v = v > 0.f ? v : NEG_ATT * v;
  atomicAdd(&den[d * heads + hh], __expf(v - mx[d * heads + hh]));
}

// one thread per (edge, channel); consecutive threads -> consecutive channels
__global__ void __launch_bounds__(256) k_edge_agg(const int* __restrict__ ei, int E, int Etot,
                                                  const float* __restrict__ ssrc,
                                                  const float* __restrict__ sdst,
                                                  const float* __restrict__ mx,
                                                  const float* __restrict__ den,
                                                  const float* __restrict__ Hn,
                                                  float* __restrict__ Out,
                                                  int heads, int cshift) {
  int idx = blockIdx.x * 256 + threadIdx.x;
  int total = Etot << 6;
  if (idx >= total) return;
  int e = idx >> 6;
  int j = idx & 63;
  int s, d; edge_ends(ei, E, e, s, d);
  int hh = j >> cshift;                         // C=16 -> 4, C=64 -> 6
  float v = ssrc[s * heads + hh] + sdst[d * heads + hh];
  v = v > 0.f ? v : NEG_ATT * v;
  float alpha = __expf(v - mx[d * heads + hh]) / den[d * heads + hh];
  atomicAdd(&Out[(size_t)d * 64 + j], Hn[(size_t)s * 64 + j] * alpha);
}

__global__ void __launch_bounds__(256) k_bias_act(float* __restrict__ A,
                                                  const float* __restrict__ b, int N) {
  int idx = blockIdx.x * 256 + threadIdx.x;
  if (idx >= N * 64) return;
  float v = A[idx] + b[idx & 63];
  A[idx] = v > 0.f ? v : NEG_OUT * v;
}

// ---------------- global max pool + FC head + softmax ----------------
__global__ void __launch_bounds__(256) k_pool(const float* __restrict__ A,
                                              const int* __restrict__ batch,
                                              float* __restrict__ pooled, int N) {
  int idx = blockIdx.x * 256 + threadIdx.x;
  if (idx >= N * 64) return;
  int n = idx >> 6, j = idx & 63;
  atomicMaxF(&pooled[batch[n] * 64 + j], A[idx]);
}

__global__ void __launch_bounds__(256) k_head(const float* __restrict__ pooled,
                                              const float* __restrict__ fcW,
                                              const float* __restrict__ fcb,
                                              float* __restrict__ out) {
  int g = threadIdx.x;                          // 256 graphs, one block
  float l0 = fcb[0], l1 = fcb[1];
  for (int k = 0; k < 64; ++k) {
    float p = pooled[g * 64 + k];
    l0 += p * fcW[k * 2 + 0];
    l1 += p * fcW[k * 2 + 1];
  }
  out[g * 2 + 0] = l0;
  out[g * 2 + 1] = l1;
  float m = fmaxf(l0, l1);
  float e0 = __expf(l0 - m), e1 = __expf(l1 - m);
  float s = e0 + e1;
  out[512 + g * 2 + 0] = e0 / s;
  out[512 + g * 2 + 1] = e1 / s;
}

// ---------------- host launch ----------------
extern "C" void kernel_launch(void* const* d_in, const int* in_sizes, int n_in,
                              void* d_out, int out_size, void* d_ws, size_t ws_size,
                              hipStream_t stream) {
  const float* x   = (const float*)d_in[0];
  const int*   ei  = (const int*)d_in[1];
  const int*   bat = (const int*)d_in[2];
  const int N    = in_sizes[2];
  const int E    = in_sizes[1] / 2;
  const int Etot = E + N;

  float* ws     = (float*)d_ws;
  float* hbuf   = ws;                           // N*64  (WMMA output h)
  float* act    = hbuf + (size_t)N * 64;        // N*64  (agg -> activation, ping)
  float* ssrc   = act  + (size_t)N * 64;        // N*4
  float* sdst   = ssrc + (size_t)N * 4;         // N*4
  float* mx     = sdst + (size_t)N * 4;         // N*4
  float* den    = mx   + (size_t)N * 4;         // N*4
  float* pooled = den  + (size_t)N * 4;         // 256*64

  struct Cfg { const float *W, *as, *ad, *b; int K, Hh, C, cshift; };
  Cfg L[4] = {
    { (const float*)d_in[3],  (const float*)d_in[4],  (const float*)d_in[5],  (const float*)d_in[6],  128, 4, 16, 4 },
    { (const float*)d_in[7],  (const float*)d_in[8],  (const float*)d_in[9],  (const float*)d_in[10],  64, 4, 16, 4 },
    { (const float*)d_in[11], (const float*)d_in[12], (const float*)d_in[13], (const float*)d_in[14],  64, 4, 16, 4 },
    { (const float*)d_in[15], (const float*)d_in[16], (const float*)d_in[17], (const float*)d_in[18],  64, 1, 64, 6 },
  };

  const float* in = x;
  for (int l = 0; l < 4; ++l) {
    const int K = L[l].K, Hh = L[l].Hh, C = L[l].C;
    k_gemm_wmma<<<(N + 31) / 32, 256, 0, stream>>>(in, L[l].W, hbuf, N, K);

    const int nh = N * Hh;
    k_attn_scores<<<(nh + 255) / 256, 256, 0, stream>>>(hbuf, L[l].as, L[l].ad,
                                                        ssrc, sdst, N, Hh, C);
    k_fill<<<(nh + 255) / 256, 256, 0, stream>>>(mx, nh, -INFINITY);
    k_fill<<<(nh + 255) / 256, 256, 0, stream>>>(den, nh, 0.f);
    k_fill<<<(N * 64 + 255) / 256, 256, 0, stream>>>(act, N * 64, 0.f);

    const int eh = Etot * Hh;
    k_edge_max<<<(eh + 255) / 256, 256, 0, stream>>>(ei, E, Etot, ssrc, sdst, mx, Hh);
    k_edge_sum<<<(eh + 255) / 256, 256, 0, stream>>>(ei, E, Etot, ssrc, sdst, mx, den, Hh);

    const int tot = Etot << 6;
    k_edge_agg<<<(tot + 255) / 256, 256, 0, stream>>>(ei, E, Etot, ssrc, sdst, mx, den,
                                                      hbuf, act, Hh, L[l].cshift);
    k_bias_act<<<(N * 64 + 255) / 256, 256, 0, stream>>>(act, L[l].b, N);
    in = act;
  }

  k_fill<<<(NGRAPH * 64 + 255) / 256, 256, 0, stream>>>(pooled, NGRAPH * 64, -INFINITY);
  k_pool<<<(N * 64 + 255) / 256, 256, 0, stream>>>(act, bat, pooled, N);
  k_head<<<1, 256, 0, stream>>>(pooled, (const float*)d_in[19], (const float*)d_in[20],
                                (float*)d_out);
}